// _VoxelPointnetSAModuleFSDistillationBase_77713138253926
// MI455X (gfx1250) — compile-verified
//
#include <hip/hip_runtime.h>
#include <hip/hip_bf16.h>

#define NPTS   16384   // N
#define NQ     4096    // NPOINT (P)
#define NS     32      // NSAMPLE (S)
#define NC     32      // feature channels C
#define NB     2       // batch
#define KPAD   36      // 3 + C padded to multiple of 4
#define R2     0.64f   // RADIUS^2

typedef float v2f __attribute__((ext_vector_type(2)));
typedef float v8f __attribute__((ext_vector_type(8)));

// ---------------------------------------------------------------------------
// Kernel 1: furthest point sampling. grid = B, block = 1024.
// Writes new_xyz (gathered coords of samples) directly to d_out[0 .. B*NQ*3).
// ---------------------------------------------------------------------------
__global__ __launch_bounds__(1024) void fps_kernel(const float* __restrict__ xyz,
                                                   float* __restrict__ newXyz)
{
    const int b   = blockIdx.x;
    const int tid = threadIdx.x;
    const int lane = tid & 31;
    const int wid  = tid >> 5;
    const float* X = xyz + (size_t)b * NPTS * 3;
    float* NX = newXyz + (size_t)b * NQ * 3;

    __shared__ float sVal[32];
    __shared__ int   sIdx[32];
    __shared__ int   sLast;

    float px[16], py[16], pz[16], dist[16];
#pragma unroll
    for (int k = 0; k < 16; ++k) {
        int i = tid + k * 1024;
        px[k] = X[i * 3 + 0];
        py[k] = X[i * 3 + 1];
        pz[k] = X[i * 3 + 2];
        dist[k] = 1e10f;
    }
    if (tid == 0) {
        NX[0] = X[0]; NX[1] = X[1]; NX[2] = X[2];
    }

    int last = 0;
    for (int j = 1; j < NQ; ++j) {
        float lx = X[last * 3 + 0];
        float ly = X[last * 3 + 1];
        float lz = X[last * 3 + 2];

        float bestV = -1.0f;
        int   bestI = 0x7fffffff;
#pragma unroll
        for (int k = 0; k < 16; ++k) {
            float dx = px[k] - lx, dy = py[k] - ly, dz = pz[k] - lz;
            float d = dx * dx + dy * dy + dz * dz;
            float dm = fminf(dist[k], d);
            dist[k] = dm;
            if (dm > bestV) { bestV = dm; bestI = tid + k * 1024; }
        }
        // wave32 reduction, ties -> smaller index
#pragma unroll
        for (int off = 16; off >= 1; off >>= 1) {
            float oV = __shfl_xor(bestV, off, 32);
            int   oI = __shfl_xor(bestI, off, 32);
            if (oV > bestV || (oV == bestV && oI < bestI)) { bestV = oV; bestI = oI; }
        }
        if (lane == 0) { sVal[wid] = bestV; sIdx[wid] = bestI; }
        __syncthreads();
        if (wid == 0) {
            bestV = sVal[lane];
            bestI = sIdx[lane];
#pragma unroll
            for (int off = 16; off >= 1; off >>= 1) {
                float oV = __shfl_xor(bestV, off, 32);
                int   oI = __shfl_xor(bestI, off, 32);
                if (oV > bestV || (oV == bestV && oI < bestI)) { bestV = oV; bestI = oI; }
            }
            if (lane == 0) {
                sLast = bestI;
                NX[j * 3 + 0] = X[bestI * 3 + 0];
                NX[j * 3 + 1] = X[bestI * 3 + 1];
                NX[j * 3 + 2] = X[bestI * 3 + 2];
            }
        }
        __syncthreads();
        last = sLast;
    }
}

// ---------------------------------------------------------------------------
// Kernel 2: transpose features [B,C,N] -> [B,N,C]
// ---------------------------------------------------------------------------
__global__ void transpose_kernel(const float* __restrict__ feat, float* __restrict__ featT)
{
    int t = blockIdx.x * blockDim.x + threadIdx.x;
    if (t >= NB * NPTS * NC) return;
    int c = t % NC;
    int n = (t / NC) % NPTS;
    int b = t / (NC * NPTS);
    featT[t] = feat[((size_t)b * NC + c) * NPTS + n];
}

// ---------------------------------------------------------------------------
// Kernel 3: ball query. One wave32 per query point. block = 256 (8 waves).
// ---------------------------------------------------------------------------
__global__ __launch_bounds__(256) void ballquery_kernel(const float* __restrict__ newXyz,
                                                        const float* __restrict__ xyz,
                                                        int* __restrict__ idxBuf,
                                                        int* __restrict__ validBuf)
{
    int lane = threadIdx.x & 31;
    int gq = blockIdx.x * 8 + (threadIdx.x >> 5);
    int b = gq >> 12;          // / NQ
    int p = gq & (NQ - 1);
    const float* Xb = xyz + (size_t)b * NPTS * 3;
    float qx = newXyz[((size_t)b * NQ + p) * 3 + 0];
    float qy = newXyz[((size_t)b * NQ + p) * 3 + 1];
    float qz = newXyz[((size_t)b * NQ + p) * 3 + 2];

    int count = 0;
    int firstIdx = -1;
    for (int base = 0; base < NPTS && count < NS; base += 32) {
        int i = base + lane;
        float dx = Xb[i * 3 + 0] - qx;
        float dy = Xb[i * 3 + 1] - qy;
        float dz = Xb[i * 3 + 2] - qz;
        float d2 = dx * dx + dy * dy + dz * dz;
        bool in = d2 < R2;
        unsigned m = (unsigned)__ballot(in);
        if (in) {
            int pos = count + __popc(m & ((1u << lane) - 1u));
            if (pos < NS) idxBuf[(size_t)gq * NS + pos] = i;
        }
        if (firstIdx < 0 && m) firstIdx = base + __builtin_ctz(m);
        count += __popc(m);
    }
    int c = count < NS ? count : NS;
    int f = firstIdx >= 0 ? firstIdx : 0;
    for (int s = c + lane; s < NS; s += 32) idxBuf[(size_t)gq * NS + s] = f;
    if (lane == 0) validBuf[gq] = (firstIdx >= 0) ? 1 : 0;
}

// ---------------------------------------------------------------------------
// Kernel 4: build grouped [B*NQ*NS, KPAD] = [rel_xyz | feat | 0-pad], masked.
// ---------------------------------------------------------------------------
__global__ void group_kernel(const float* __restrict__ xyz,
                             const float* __restrict__ newXyz,
                             const float* __restrict__ featT,
                             const int* __restrict__ idxBuf,
                             const int* __restrict__ validBuf,
                             float* __restrict__ grouped)
{
    long long t = (long long)blockIdx.x * blockDim.x + threadIdx.x;
    const long long total = (long long)NB * NQ * NS * KPAD;
    if (t >= total) return;
    int c = (int)(t % KPAD);
    int r = (int)(t / KPAD);              // row in [0, B*NQ*NS)
    int idx = idxBuf[r];
    float valid = (float)validBuf[r / NS];
    int b = r / (NQ * NS);
    int p = (r / NS) & (NQ - 1);
    float v;
    if (c < 3) {
        v = xyz[((size_t)b * NPTS + idx) * 3 + c] - newXyz[((size_t)b * NQ + p) * 3 + c];
    } else if (c < 3 + NC) {
        v = featT[((size_t)b * NPTS + idx) * NC + (c - 3)];
    } else {
        v = 0.0f;
    }
    grouped[(size_t)r * KPAD + c] = v * valid;
}

// ---------------------------------------------------------------------------
// Kernel 5: fused 2-layer MLP + max-pool with V_WMMA_F32_16X16X4_F32.
// block = 64 (2 waves), each wave owns one (b,p) group of 32 rows.
// ---------------------------------------------------------------------------
__global__ __launch_bounds__(64) void mlp_kernel(const float* __restrict__ grouped,
                                                 const float* __restrict__ W1,
                                                 const float* __restrict__ b1,
                                                 const float* __restrict__ W2,
                                                 const float* __restrict__ b2,
                                                 float* __restrict__ outFeat)
{
    __shared__ float lW1[KPAD * 64];     //  9216 B (row 35 zero-padded)
    __shared__ float lW2[64 * 128];      // 32768 B
    __shared__ float lB1[64];
    __shared__ float lB2[128];
    __shared__ float lH[2][32 * 66];     // per-wave layer-1 output, stride 66

    const int tid = threadIdx.x;
    for (int i = tid; i < KPAD * 64; i += 64) {
        int row = i >> 6;
        lW1[i] = (row < 35) ? W1[i] : 0.0f;
    }
    for (int i = tid; i < 64 * 128; i += 64) lW2[i] = W2[i];
    lB1[tid] = b1[tid];
    lB2[tid] = b2[tid];
    lB2[tid + 64] = b2[tid + 64];
    __syncthreads();

    const int wid  = tid >> 5;
    const int lane = tid & 31;
    const int lm = lane & 15;
    const int kh = lane >> 4;
    const int g  = blockIdx.x * 2 + wid;      // group id in [0, B*NQ)
    const int b  = g >> 12;
    const int p  = g & (NQ - 1);

    const float* Xg = grouped + (size_t)g * NS * KPAD;
    float* H = &lH[wid][0];

    // ---- Layer 1: [32 x 36] x [36 x 64], K-steps of 4 ----
    for (int t = 0; t < 2; ++t) {
        v2f a[9];
#pragma unroll
        for (int k = 0; k < 9; ++k)
            a[k] = *(const v2f*)(Xg + (t * 16 + lm) * KPAD + 4 * k + 2 * kh);
#pragma unroll
        for (int n = 0; n < 4; ++n) {
            v8f acc = {};
#pragma unroll
            for (int k = 0; k < 9; ++k) {
                v2f bb;
                bb.x = lW1[(4 * k + 2 * kh) * 64 + n * 16 + lm];
                bb.y = lW1[(4 * k + 2 * kh + 1) * 64 + n * 16 + lm];
                acc = __builtin_amdgcn_wmma_f32_16x16x4_f32(
                        false, a[k], false, bb, (short)0, acc, false, false);
            }
            float bias = lB1[n * 16 + lm];
#pragma unroll
            for (int v = 0; v < 8; ++v) {
                float h = fmaxf(acc[v] + bias, 0.0f);
                H[(t * 16 + v + 8 * kh) * 66 + n * 16 + lm] = h;
            }
        }
    }
    // same-wave DS ops are in-order: layer-2 loads see layer-1 stores.

    // ---- Layer 2: [32 x 64] x [64 x 128] + ReLU + max over the 32 rows ----
    float* outp = outFeat + (size_t)b * 128 * NQ + p;
#pragma unroll 1
    for (int n = 0; n < 8; ++n) {
        float pool = 0.0f;   // all values are post-ReLU >= 0
        for (int t = 0; t < 2; ++t) {
            v8f acc = {};
#pragma unroll
            for (int k = 0; k < 16; ++k) {
                v2f a2 = *(const v2f*)(H + (t * 16 + lm) * 66 + 4 * k + 2 * kh);
                v2f bb;
                bb.x = lW2[(4 * k + 2 * kh) * 128 + n * 16 + lm];
                bb.y = lW2[(4 * k + 2 * kh + 1) * 128 + n * 16 + lm];
                acc = __builtin_amdgcn_wmma_f32_16x16x4_f32(
                        false, a2, false, bb, (short)0, acc, false, false);
            }
            float bias = lB2[n * 16 + lm];
            float mv = 0.0f;
#pragma unroll
            for (int v = 0; v < 8; ++v) mv = fmaxf(mv, fmaxf(acc[v] + bias, 0.0f));
            mv = fmaxf(mv, __shfl_xor(mv, 16, 32));   // combine the two row-halves
            pool = fmaxf(pool, mv);                   // combine the two M-tiles
        }
        if (lane < 16) outp[(size_t)(n * 16 + lm) * NQ] = pool;
    }
}

// ---------------------------------------------------------------------------
// Host launcher
// ---------------------------------------------------------------------------
extern "C" void kernel_launch(void* const* d_in, const int* in_sizes, int n_in,
                              void* d_out, int out_size, void* d_ws, size_t ws_size,
                              hipStream_t stream)
{
    const float* xyz  = (const float*)d_in[0];   // [B, N, 3]
    const float* feat = (const float*)d_in[1];   // [B, C, N]
    const float* W1   = (const float*)d_in[2];   // [35, 64]
    const float* b1   = (const float*)d_in[3];   // [64]
    const float* W2   = (const float*)d_in[4];   // [64, 128]
    const float* b2   = (const float*)d_in[5];   // [128]

    float* out     = (float*)d_out;
    float* newXyz  = out;                        // [B, NQ, 3]
    float* newFeat = out + (size_t)NB * NQ * 3;  // [B, 128, NQ]

    char* ws = (char*)d_ws;
    int*   idxBuf   = (int*)(ws);                               // B*NQ*NS ints  = 1 MB
    int*   validBuf = (int*)(ws + 1048576);                     // B*NQ ints     = 32 KB
    float* featT    = (float*)(ws + 1081344);                   // B*N*C floats  = 4 MB
    float* grouped  = (float*)(ws + 5275648);                   // B*NQ*NS*KPAD  = 37.75 MB

    // 1. FPS -> new_xyz
    fps_kernel<<<NB, 1024, 0, stream>>>(xyz, newXyz);

    // 2. feature transpose
    {
        int total = NB * NPTS * NC;
        transpose_kernel<<<(total + 255) / 256, 256, 0, stream>>>(feat, featT);
    }

    // 3. ball query
    ballquery_kernel<<<(NB * NQ) / 8, 256, 0, stream>>>(newXyz, xyz, idxBuf, validBuf);

    // 4. gather / group (zero-masked, K padded to 36)
    {
        long long total = (long long)NB * NQ * NS * KPAD;
        int blocks = (int)((total + 255) / 256);
        group_kernel<<<blocks, 256, 0, stream>>>(xyz, newXyz, featT, idxBuf, validBuf, grouped);
    }

    // 5. WMMA MLP + max-pool -> new_features
    mlp_kernel<<<(NB * NQ) / 2, 64, 0, stream>>>(grouped, W1, b1, W2, b2, newFeat);

    (void)in_sizes; (void)n_in; (void)out_size; (void)ws_size;
}